// quantiser_60387240182069
// MI455X (gfx1250) — compile-verified
//
#include <hip/hip_runtime.h>
#include <math.h>

typedef __attribute__((ext_vector_type(2))) float v2f;
typedef __attribute__((ext_vector_type(8))) float v8f;

// Problem sizes (fixed by the reference)
constexpr int Bn = 256;   // batch
constexpr int Dn = 512;   // feature dim
constexpr int Gn = 64;    // groups
constexpr int Nn = 8;     // codes per group
constexpr int Kn = Gn * Nn; // 512 codebook entries

// ---------------------------------------------------------------------------
// K0: input_sig = exp(input_logsig)
// ---------------------------------------------------------------------------
__global__ __launch_bounds__(256) void prep_exp_kernel(
    const float* __restrict__ input_logsig, float* __restrict__ sig) {
  int t = blockIdx.x * 256 + threadIdx.x;
  if (t < Bn * Dn) sig[t] = __expf(input_logsig[t]);
}

// ---------------------------------------------------------------------------
// K1: row norms.  t<256 -> norm_b[b] = sum_d mu^2+sig^2 ; t in [256,768) ->
//     norm_k[k] = sum over on_states[k,:,:]^2 (covers both mu and sig planes)
// ---------------------------------------------------------------------------
__global__ __launch_bounds__(256) void norms_kernel(
    const float* __restrict__ input_mu, const float* __restrict__ sig,
    const float* __restrict__ on_states,
    float* __restrict__ norm_b, float* __restrict__ norm_k) {
  int t = blockIdx.x * 256 + threadIdx.x;
  if (t < Bn) {
    const float* m = input_mu + t * Dn;
    const float* s = sig + t * Dn;
    float acc = 0.f;
    for (int d = 0; d < Dn; ++d) acc = fmaf(m[d], m[d], fmaf(s[d], s[d], acc));
    norm_b[t] = acc;
  } else if (t < Bn + Kn) {
    int k = t - Bn;
    const float* o = on_states + (size_t)k * (Dn * 2);
    float acc = 0.f;
    for (int j = 0; j < Dn * 2; ++j) acc = fmaf(o[j], o[j], acc);
    norm_k[k] = acc;
  }
}

// ---------------------------------------------------------------------------
// K2: d_kb[k,b] = norm_k + norm_b - 2*(mus@muT + sigs@sigT)  via f32 WMMA.
// One wave per 16x16 (k,b) tile; K-loop over D in steps of 4.
// on_states interleave [k][d][2] -> one float4 = (mu_d, sig_d, mu_{d+1}, sig_{d+1})
// feeds both the mu and sig A-fragments.
// d_kb written flat at k*256+b == reference's reshape(256,512) view.
// ---------------------------------------------------------------------------
__global__ __launch_bounds__(256) void w2_dist_kernel(
    const float* __restrict__ on_states, const float* __restrict__ input_mu,
    const float* __restrict__ sig, const float* __restrict__ norm_k,
    const float* __restrict__ norm_b, float* __restrict__ dists) {
  const int lane = threadIdx.x & 31;
  const int wave = threadIdx.x >> 5;
  const int tile = blockIdx.x * 8 + wave;   // 512 tiles = 32 ktiles * 16 btiles
  const int kt = tile >> 4;
  const int bt = tile & 15;
  const int m  = lane & 15;                 // A row index / B col index
  const int kb = (lane >> 4) << 1;          // K sub-index base: 0 or 2
  const int krow = kt * 16 + m;             // absolute codebook row
  const int bcol = bt * 16 + m;             // absolute batch col

  const float4* on4 = (const float4*)on_states;  // [K][256] float4
  v8f cmu = {};
  v8f csg = {};
#pragma unroll 4
  for (int d0 = 0; d0 < Dn; d0 += 4) {
    const int dd = d0 + kb;
    const float4 av = on4[krow * 256 + (dd >> 1)];
    v2f amu; amu[0] = av.x; amu[1] = av.z;
    v2f asg; asg[0] = av.y; asg[1] = av.w;
    const float2 bm = *(const float2*)(input_mu + bcol * Dn + dd);
    const float2 bs = *(const float2*)(sig + bcol * Dn + dd);
    v2f bmu; bmu[0] = bm.x; bmu[1] = bm.y;
    v2f bsg; bsg[0] = bs.x; bsg[1] = bs.y;
    cmu = __builtin_amdgcn_wmma_f32_16x16x4_f32(false, amu, false, bmu,
                                                (short)0, cmu, false, false);
    csg = __builtin_amdgcn_wmma_f32_16x16x4_f32(false, asg, false, bsg,
                                                (short)0, csg, false, false);
  }
  // C layout: VGPR r holds row (r + 8*(lane>=16)), col = lane&15
  const int colb  = bt * 16 + (lane & 15);
  const int rbase = kt * 16 + ((lane >> 4) << 3);
  const float nb = norm_b[colb];
#pragma unroll
  for (int r = 0; r < 8; ++r) {
    const int kr = rbase + r;
    dists[kr * Bn + colb] = norm_k[kr] + nb - 2.0f * (cmu[r] + csg[r]);
  }
}

// ---------------------------------------------------------------------------
// K3: per (i,g): softmax over the 8 codes; emit softs and dists_bg.
// ---------------------------------------------------------------------------
__global__ __launch_bounds__(256) void soft_kernel(
    const float* __restrict__ dists, float* __restrict__ softs,
    float* __restrict__ dists_bg, float* __restrict__ out_dists_bg) {
  int t = blockIdx.x * 256 + threadIdx.x;   // 16384 = 256*64
  if (t >= Bn * Gn) return;
  const int i = t >> 6, g = t & 63;
  const float* dp = dists + i * 512 + g * 8;
  float d[8], mn = dp[0];
#pragma unroll
  for (int n = 0; n < 8; ++n) { d[n] = dp[n]; mn = fminf(mn, d[n]); }
  float e[8], s = 0.f;
#pragma unroll
  for (int n = 0; n < 8; ++n) { e[n] = __expf(mn - d[n]); s += e[n]; }
  const float inv = __frcp_rn(s);
  float acc = 0.f;
  float* sp = softs + i * 512 + g * 8;
#pragma unroll
  for (int n = 0; n < 8; ++n) { float w = e[n] * inv; sp[n] = w; acc = fmaf(w, d[n], acc); }
  dists_bg[t] = acc;
  out_dists_bg[t] = acc;
}

// ---------------------------------------------------------------------------
// K4: ps = softmax(-dists_bg, axis=0): one block per g, 256 threads (one per b)
// ---------------------------------------------------------------------------
__global__ __launch_bounds__(256) void ps_kernel(
    const float* __restrict__ dists_bg, float* __restrict__ ps) {
  __shared__ float red[256];
  const int g = blockIdx.x;
  const int b = threadIdx.x;
  const float v = -dists_bg[b * Gn + g];
  red[b] = v; __syncthreads();
  for (int s = 128; s > 0; s >>= 1) {
    if (b < s) red[b] = fmaxf(red[b], red[b + s]);
    __syncthreads();
  }
  const float mx = red[0];
  __syncthreads();
  const float e = __expf(v - mx);
  red[b] = e; __syncthreads();
  for (int s = 128; s > 0; s >>= 1) {
    if (b < s) red[b] += red[b + s];
    __syncthreads();
  }
  ps[b * Gn + g] = e * __frcp_rn(red[0]);
}

// ---------------------------------------------------------------------------
// K5: dist[b] = sum_g dists_bg[b,g]*ps[b,g]
// ---------------------------------------------------------------------------
__global__ __launch_bounds__(256) void final_dist_kernel(
    const float* __restrict__ dists_bg, const float* __restrict__ ps,
    float* __restrict__ out_dist) {
  int b = blockIdx.x * 256 + threadIdx.x;
  if (b >= Bn) return;
  float acc = 0.f;
  for (int g = 0; g < Gn; ++g) acc = fmaf(dists_bg[b * Gn + g], ps[b * Gn + g], acc);
  out_dist[b] = acc;
}

// ---------------------------------------------------------------------------
// K6: the bandwidth-dominant kernel (268 MB of stores).
// block = (x, d-chunk of 64). Build states_x[e][64g][64d] in LDS, then
// 8 waves run ps(256x64) @ states(64x64) with f32 WMMA and stream results out.
// ---------------------------------------------------------------------------
__global__ __launch_bounds__(256) void quant_kernel(
    const float* __restrict__ on_states, const float* __restrict__ softs,
    const float* __restrict__ ps, float* __restrict__ out) {
  constexpr int DCH = 64;
  constexpr int SST = 65;                    // padded row stride (bank-conflict free)
  __shared__ float s_softs[Gn * Nn];         // 2 KB
  __shared__ float s_states[2 * Gn * SST];   // ~33 KB
  const int x = blockIdx.y;
  const int dbase = blockIdx.x * DCH;

  for (int i = threadIdx.x; i < Gn * Nn; i += 256) s_softs[i] = softs[x * 512 + i];
  __syncthreads();

  // states[e][g][d] = sum_n softs[x,g,n] * on_states[g*8+n, dbase+d, e]
  for (int idx = threadIdx.x; idx < Gn * DCH; idx += 256) {
    const int g = idx >> 6, d = idx & 63;
    float am = 0.f, as = 0.f;
#pragma unroll
    for (int n = 0; n < 8; ++n) {
      const float w = s_softs[g * 8 + n];
      const float2 ms = *(const float2*)(on_states + (size_t)(g * 8 + n) * (Dn * 2)
                                         + (size_t)(dbase + d) * 2);
      am = fmaf(w, ms.x, am);
      as = fmaf(w, ms.y, as);
    }
    s_states[0 * Gn * SST + g * SST + d] = am;
    s_states[1 * Gn * SST + g * SST + d] = as;
  }
  __syncthreads();

  const int lane = threadIdx.x & 31;
  const int wave = threadIdx.x >> 5;
  const int m  = lane & 15;
  const int kb = (lane >> 4) << 1;

  for (int bt = wave; bt < 16; bt += 8) {     // 16 batch tiles over 8 waves
    // Preload all A fragments for this row of ps (reused across e and dtiles)
    v2f afr[16];
    const float* prow = ps + (bt * 16 + m) * Gn;
#pragma unroll
    for (int kc = 0; kc < 16; ++kc) {
      afr[kc][0] = prow[kc * 4 + kb];
      afr[kc][1] = prow[kc * 4 + kb + 1];
    }
#pragma unroll
    for (int e = 0; e < 2; ++e) {
      const float* st = s_states + e * Gn * SST;
#pragma unroll
      for (int dt = 0; dt < DCH / 16; ++dt) {
        v8f c = {};
#pragma unroll
        for (int kc = 0; kc < 16; ++kc) {
          const float* srow = st + (kc * 4 + kb) * SST + dt * 16 + m;
          v2f bfr; bfr[0] = srow[0]; bfr[1] = srow[SST];
          c = __builtin_amdgcn_wmma_f32_16x16x4_f32(false, afr[kc], false, bfr,
                                                    (short)0, c, false, false);
        }
        float* o = out + (size_t)e * Bn * Bn * Dn + (size_t)x * Bn * Dn
                 + dbase + dt * 16 + (lane & 15);
        const int rbase = bt * 16 + ((lane >> 4) << 3);
#pragma unroll
        for (int r = 0; r < 8; ++r)
          o[(size_t)(rbase + r) * Dn] = c[r];
      }
    }
  }
}

// ---------------------------------------------------------------------------
extern "C" void kernel_launch(void* const* d_in, const int* in_sizes, int n_in,
                              void* d_out, int out_size, void* d_ws, size_t ws_size,
                              hipStream_t stream) {
  const float* input_mu     = (const float*)d_in[0];   // [256,512]
  const float* input_logsig = (const float*)d_in[1];   // [256,512]
  const float* on_states    = (const float*)d_in[2];   // [512,512,2]
  float* out = (float*)d_out;
  float* ws  = (float*)d_ws;

  // workspace layout (floats): ~1.7 MB total
  float* w_sig   = ws;               // 131072
  float* w_dists = ws + 131072;      // 131072  (d_kb flat == dists[256][512])
  float* w_softs = ws + 262144;      // 131072
  float* w_dbg   = ws + 393216;      // 16384
  float* w_ps    = ws + 409600;      // 16384
  float* w_nb    = ws + 425984;      // 256
  float* w_nk    = ws + 426240;      // 512

  float* out_dbg  = out + (size_t)2 * Bn * Bn * Dn;    // dists_bg [256,64]
  float* out_dist = out_dbg + (size_t)Bn * Gn;         // dist [256]

  prep_exp_kernel<<<(Bn * Dn + 255) / 256, 256, 0, stream>>>(input_logsig, w_sig);
  norms_kernel<<<3, 256, 0, stream>>>(input_mu, w_sig, on_states, w_nb, w_nk);
  w2_dist_kernel<<<64, 256, 0, stream>>>(on_states, input_mu, w_sig, w_nk, w_nb, w_dists);
  soft_kernel<<<(Bn * Gn + 255) / 256, 256, 0, stream>>>(w_dists, w_softs, w_dbg, out_dbg);
  ps_kernel<<<Gn, 256, 0, stream>>>(w_dbg, w_ps);
  final_dist_kernel<<<1, 256, 0, stream>>>(w_dbg, w_ps, out_dist);
  quant_kernel<<<dim3(Dn / 64, Bn), 256, 0, stream>>>(on_states, w_softs, w_ps, out);
}